// EdgeConv_20289425506394
// MI455X (gfx1250) — compile-verified
//
#include <hip/hip_runtime.h>

#define DEVFN __device__ __forceinline__

typedef float v2f __attribute__((ext_vector_type(2)));
typedef float v8f __attribute__((ext_vector_type(8)));

constexpr int   kN = 100000;
constexpr int   kE = 1600000;
constexpr int   kB = 512;
constexpr float kEPS = 1e-5f;
constexpr int   GRID_CONV = 2048;   // fixed -> deterministic partial-sum layout

// ---- fp32 WMMA: D(16x16) = A(16x4) * B(4x16) + C --------------------------
DEVFN v8f wmma4(v2f a, v2f b, v8f c) {
  return __builtin_amdgcn_wmma_f32_16x16x4_f32(false, a, false, b, (short)0, c,
                                               false, false);
}

// feature k of edge-row l: concat([xi, xj - xi]) (xj stored 16B-XOR-swizzled
// for IN_F==64 so async-staged chunks read back conflict-free)
template <int IN_F>
DEVFN float featval(const float* xi, const float* xj, int l, int k) {
  if constexpr (IN_F == 64) {
    if (k < 64) return xi[k];
    int c = k - 64;
    return xj[l * 64 + (((c >> 2) ^ l) << 2) + (c & 3)] - xi[c];
  } else {
    if (k < IN_F) return xi[k];
    if (k < 2 * IN_F) return xj[l * IN_F + (k - IN_F)] - xi[k - IN_F];
    return 0.0f;
  }
}

// m1[16 x 64] += A[16 x KPAD] @ W1 (pair-packed v2f LDS), 4 N-tiles
template <int IN_F, int KPAD>
DEVFN void m1_tile(const float* xi, const float* xj, const v2f* W1p, int l,
                   int hi, v8f acc[4]) {
  for (int k0 = 0; k0 < KPAD; k0 += 4) {
    int kk = k0 + 2 * hi;  // even
    v2f a;
    a.x = featval<IN_F>(xi, xj, l, kk);
    a.y = featval<IN_F>(xi, xj, l, kk + 1);
    const v2f* wrow = &W1p[(kk >> 1) * 64 + l];
#pragma unroll
    for (int nt = 0; nt < 4; ++nt) {
      v2f b = wrow[nt * 16];  // single aligned ds_load_b64
      acc[nt] = wmma4(a, b, acc[nt]);
    }
  }
}

// stage xi / xj tile for node into per-wave LDS.
// IN_F==64: async DMA (GLOBAL_LOAD_ASYNC_TO_LDS_B128) with 16B XOR swizzle.
template <int IN_F>
DEVFN void stage_tile(const float* __restrict__ feat,
                      const int* __restrict__ src, int node, int lane,
                      float* xi, float* xj) {
  if constexpr (IN_F == 64) {
    // previous iteration's DS reads/writes of this LDS region must retire
    // before unordered async writes may land.
    asm volatile("s_wait_dscnt 0x0" ::: "memory");
    if (lane < 16)
      *(float4*)&xi[lane * 4] =
          *(const float4*)&feat[(size_t)node * 64 + lane * 4];
    unsigned base = (unsigned)(size_t)xj;  // LDS generic addr low 32 = offset
#pragma unroll
    for (int it = 0; it < 8; ++it) {
      int idx = lane + 32 * it;
      int r = idx >> 4, c4 = idx & 15;
      const float* gp = feat + (size_t)src[node * 16 + r] * 64 + c4 * 4;
      unsigned loff = base + (unsigned)(r * 256 + ((c4 ^ r) << 4));
      asm volatile("global_load_async_to_lds_b128 %0, %1, off" ::"v"(loff),
                   "v"(gp)
                   : "memory");
    }
    asm volatile("s_wait_asynccnt 0x0" ::: "memory");
  } else {
    for (int i = lane; i < IN_F; i += 32) xi[i] = feat[(size_t)node * IN_F + i];
    for (int i = lane; i < 16 * IN_F; i += 32) {
      int r = i / IN_F, c = i - r * IN_F;
      xj[r * IN_F + c] = feat[(size_t)src[node * 16 + r] * IN_F + c];
    }
  }
}

// ---- Pass A: accumulate per-channel sum / sumsq of m1 (bias-free) ---------
template <int IN_F, int KPAD>
__global__ __launch_bounds__(64) void conv_passA(const float* __restrict__ feat,
                                                 const int* __restrict__ src,
                                                 const float* __restrict__ W1,
                                                 float* __restrict__ part) {
  __shared__ v2f W1p[(KPAD / 2) * 64];
  __shared__ __align__(16) float xis[2][64];
  __shared__ __align__(16) float xjm[2][1056];
  __shared__ float wsum[2][64];
  __shared__ float wsq[2][64];

  int tid = threadIdx.x, wave = tid >> 5, lane = tid & 31;
  int l = lane & 15, hi = lane >> 4;

  for (int i = tid; i < KPAD * 64; i += 64) {
    int r = i >> 6, c = i & 63;
    W1p[(r >> 1) * 64 + c][r & 1] = (r < 2 * IN_F) ? W1[i] : 0.0f;
  }
  __syncthreads();

  float esum[4] = {0, 0, 0, 0}, esq[4] = {0, 0, 0, 0};
  for (int n = blockIdx.x * 2 + wave; n < kN; n += GRID_CONV * 2) {
    int nn = n + GRID_CONV * 2;
    if (nn < kN) __builtin_prefetch(&feat[(size_t)nn * IN_F], 0, 0);
    stage_tile<IN_F>(feat, src, n, lane, xis[wave], xjm[wave]);
    v8f acc[4];
#pragma unroll
    for (int nt = 0; nt < 4; ++nt)
#pragma unroll
      for (int j = 0; j < 8; ++j) acc[nt][j] = 0.0f;
    m1_tile<IN_F, KPAD>(xis[wave], xjm[wave], W1p, l, hi, acc);
#pragma unroll
    for (int nt = 0; nt < 4; ++nt) {
      float s = 0, q = 0;
#pragma unroll
      for (int j = 0; j < 8; ++j) {
        float v = acc[nt][j];
        s += v;
        q += v * v;
      }
      esum[nt] += s;
      esq[nt] += q;
    }
  }
#pragma unroll
  for (int nt = 0; nt < 4; ++nt) {
    esum[nt] += __shfl_xor(esum[nt], 16);
    esq[nt] += __shfl_xor(esq[nt], 16);
  }
  if (hi == 0) {
#pragma unroll
    for (int nt = 0; nt < 4; ++nt) {
      wsum[wave][nt * 16 + l] = esum[nt];
      wsq[wave][nt * 16 + l] = esq[nt];
    }
  }
  __syncthreads();
  if (tid < 64) {
    part[(size_t)blockIdx.x * 128 + tid] = wsum[0][tid] + wsum[1][tid];
    part[(size_t)blockIdx.x * 128 + 64 + tid] = wsq[0][tid] + wsq[1][tid];
  }
}

// ---- deterministic stats reduction -> scale/shift -------------------------
__global__ void reduce_stats(const float* __restrict__ part, int G,
                             float invCount, const float* __restrict__ bias,
                             const float* __restrict__ gamma,
                             const float* __restrict__ beta,
                             float* __restrict__ ss) {
  int c = threadIdx.x;  // 64 threads
  float s = 0, q = 0;
  for (int g = 0; g < G; ++g) {
    s += part[(size_t)g * 128 + c];
    q += part[(size_t)g * 128 + 64 + c];
  }
  float mean = s * invCount;
  float var = q * invCount - mean * mean;  // bias shifts mean only, not var
  if (bias) mean += bias[c];
  float sc = gamma[c] * rsqrtf(var + kEPS);
  ss[c] = sc;
  ss[64 + c] = beta[c] - mean * sc;
}

// ---- Pass B: recompute m1, BN+ReLU, m2 = m1n@W2+b2, segment-max over 16 ---
template <int IN_F, int KPAD>
__global__ __launch_bounds__(64) void conv_passB(
    const float* __restrict__ feat, const int* __restrict__ src,
    const float* __restrict__ W1, const float* __restrict__ b1,
    const float* __restrict__ W2, const float* __restrict__ b2,
    const float* __restrict__ ss, float* __restrict__ hagg,
    float* __restrict__ part) {
  __shared__ v2f W1p[(KPAD / 2) * 64];
  __shared__ v2f W2p[32 * 64];
  __shared__ __align__(16) float xis[2][64];
  __shared__ __align__(16) float xjm[2][1056];  // xj tile / m1n (stride 66)
  __shared__ float wsum[2][64];
  __shared__ float wsq[2][64];

  int tid = threadIdx.x, wave = tid >> 5, lane = tid & 31;
  int l = lane & 15, hi = lane >> 4;

  for (int i = tid; i < KPAD * 64; i += 64) {
    int r = i >> 6, c = i & 63;
    W1p[(r >> 1) * 64 + c][r & 1] = (r < 2 * IN_F) ? W1[i] : 0.0f;
  }
  for (int i = tid; i < 64 * 64; i += 64) {
    int r = i >> 6, c = i & 63;
    W2p[(r >> 1) * 64 + c][r & 1] = W2[i];
  }
  __syncthreads();

  float bias1[4], bias2[4], sc[4], sh[4];
#pragma unroll
  for (int nt = 0; nt < 4; ++nt) {
    int ch = nt * 16 + l;
    bias1[nt] = b1[ch];
    bias2[nt] = b2[ch];
    sc[nt] = ss[ch];
    sh[nt] = ss[64 + ch];
  }

  float nsum[4] = {0, 0, 0, 0}, nsq[4] = {0, 0, 0, 0};
  for (int n = blockIdx.x * 2 + wave; n < kN; n += GRID_CONV * 2) {
    int nn = n + GRID_CONV * 2;
    if (nn < kN) __builtin_prefetch(&feat[(size_t)nn * IN_F], 0, 0);
    stage_tile<IN_F>(feat, src, n, lane, xis[wave], xjm[wave]);
    v8f acc[4];
#pragma unroll
    for (int nt = 0; nt < 4; ++nt)
#pragma unroll
      for (int j = 0; j < 8; ++j) acc[nt][j] = bias1[nt];
    m1_tile<IN_F, KPAD>(xis[wave], xjm[wave], W1p, l, hi, acc);

    // BN + ReLU, store m1n[16 x 64] to LDS (stride 66: even -> b64 A loads)
    float* m1s = xjm[wave];
#pragma unroll
    for (int nt = 0; nt < 4; ++nt)
#pragma unroll
      for (int j = 0; j < 8; ++j) {
        float v = fmaxf(acc[nt][j] * sc[nt] + sh[nt], 0.0f);
        m1s[(hi * 8 + j) * 66 + nt * 16 + l] = v;
      }

    // m2 = m1n @ W2 + b2  (K = 64)
    v8f acc2[4];
#pragma unroll
    for (int nt = 0; nt < 4; ++nt)
#pragma unroll
      for (int j = 0; j < 8; ++j) acc2[nt][j] = bias2[nt];
    for (int k0 = 0; k0 < 64; k0 += 4) {
      int kk = k0 + 2 * hi;  // even; l*66 + kk even -> 8B aligned
      v2f a = *(const v2f*)__builtin_assume_aligned(&m1s[l * 66 + kk], 8);
      const v2f* wrow = &W2p[(kk >> 1) * 64 + l];
#pragma unroll
      for (int nt = 0; nt < 4; ++nt) {
        v2f b = wrow[nt * 16];
        acc2[nt] = wmma4(a, b, acc2[nt]);
      }
    }

    // segment-max over the 16 edges of node n (all rows of the tile)
#pragma unroll
    for (int nt = 0; nt < 4; ++nt) {
      float m = acc2[nt][0];
#pragma unroll
      for (int j = 1; j < 8; ++j) m = fmaxf(m, acc2[nt][j]);
      m = fmaxf(m, __shfl_xor(m, 16));
      if (hi == 0) {
        hagg[(size_t)n * 64 + nt * 16 + l] = m;
        nsum[nt] += m;
        nsq[nt] += m * m;
      }
    }
  }
  if (hi == 0) {
#pragma unroll
    for (int nt = 0; nt < 4; ++nt) {
      wsum[wave][nt * 16 + l] = nsum[nt];
      wsq[wave][nt * 16 + l] = nsq[nt];
    }
  }
  __syncthreads();
  if (tid < 64) {
    part[(size_t)blockIdx.x * 128 + tid] = wsum[0][tid] + wsum[1][tid];
    part[(size_t)blockIdx.x * 128 + 64 + tid] = wsq[0][tid] + wsq[1][tid];
  }
}

// ---- inter-layer BN + ReLU; write h and the JK cat column -----------------
__global__ void node_bn_relu(const float* __restrict__ hagg,
                             const float* __restrict__ ss, float* __restrict__ h,
                             float* __restrict__ xs, int col0) {
  int i = blockIdx.x * 256 + threadIdx.x;
  if (i >= kN * 64) return;
  int c = i & 63, n = i >> 6;
  float v = fmaxf(hagg[i] * ss[c] + ss[64 + c], 0.0f);
  h[i] = v;
  xs[(size_t)n * 192 + col0 + c] = v;
}

// ---- JK: hjk[N,64] = xs[N,192] @ Wjk + bjk (WMMA, K=192) ------------------
__global__ __launch_bounds__(64) void jk_kernel(const float* __restrict__ xs,
                                                const float* __restrict__ Wjk,
                                                const float* __restrict__ bjk,
                                                float* __restrict__ hjk) {
  __shared__ v2f Wp[96 * 64];
  int tid = threadIdx.x, wave = tid >> 5, lane = tid & 31;
  int l = lane & 15, hi = lane >> 4;
  for (int i = tid; i < 192 * 64; i += 64) {
    int r = i >> 6, c = i & 63;
    Wp[(r >> 1) * 64 + c][r & 1] = Wjk[i];
  }
  __syncthreads();
  float bj[4];
#pragma unroll
  for (int nt = 0; nt < 4; ++nt) bj[nt] = bjk[nt * 16 + l];

  for (int t = blockIdx.x * 2 + wave; t < kN / 16; t += 512 * 2) {
    v8f acc[4];
#pragma unroll
    for (int nt = 0; nt < 4; ++nt)
#pragma unroll
      for (int j = 0; j < 8; ++j) acc[nt][j] = bj[nt];
    const float* Arow = xs + (size_t)(t * 16 + l) * 192;
    for (int k0 = 0; k0 < 192; k0 += 4) {
      int kk = k0 + 2 * hi;
      v2f a;
      a.x = Arow[kk];
      a.y = Arow[kk + 1];
      const v2f* wrow = &Wp[(kk >> 1) * 64 + l];
#pragma unroll
      for (int nt = 0; nt < 4; ++nt) {
        v2f b = wrow[nt * 16];
        acc[nt] = wmma4(a, b, acc[nt]);
      }
    }
#pragma unroll
    for (int nt = 0; nt < 4; ++nt)
#pragma unroll
      for (int j = 0; j < 8; ++j)
        hjk[(size_t)(t * 16 + hi * 8 + j) * 64 + nt * 16 + l] = acc[nt][j];
  }
}

// ---- global_add_pool over sorted batch (deterministic, 1 block / segment) -
__global__ void pool_kernel(const float* __restrict__ hjk,
                            const int* __restrict__ batch,
                            float* __restrict__ g) {
  int b = blockIdx.x, c = threadIdx.x;  // 64 threads
  int lo, hi;
  {
    int a = 0, z = kN;
    while (a < z) { int m = (a + z) >> 1; if (batch[m] < b) a = m + 1; else z = m; }
    lo = a;
  }
  {
    int a = 0, z = kN;
    while (a < z) { int m = (a + z) >> 1; if (batch[m] < b + 1) a = m + 1; else z = m; }
    hi = a;
  }
  float s = 0;
  for (int n = lo; n < hi; ++n) s += hjk[(size_t)n * 64 + c];
  g[(size_t)b * 64 + c] = s;
}

// ---- classifier (tiny: 8 MFLOP) -------------------------------------------
__global__ void cls1(const float* __restrict__ g, const float* __restrict__ W,
                     const float* __restrict__ b, float* __restrict__ z) {
  int i = blockIdx.x * 256 + threadIdx.x;  // kB*128
  if (i >= kB * 128) return;
  int c = i & 127, r = i >> 7;
  float s = b[c];
  for (int k = 0; k < 64; ++k) s += g[r * 64 + k] * W[k * 128 + c];
  z[i] = s;
}
__global__ void cls_stats(const float* __restrict__ z,
                          const float* __restrict__ gamma,
                          const float* __restrict__ beta,
                          float* __restrict__ ss) {
  int c = threadIdx.x;  // 128 threads
  float s = 0, q = 0;
  for (int r = 0; r < kB; ++r) { float v = z[r * 128 + c]; s += v; q += v * v; }
  float mean = s / kB, var = q / kB - mean * mean;
  float sc = gamma[c] * rsqrtf(var + kEPS);
  ss[c] = sc;
  ss[128 + c] = beta[c] - mean * sc;
}
__global__ void cls2(const float* __restrict__ z, const float* __restrict__ ss,
                     const float* __restrict__ W, const float* __restrict__ b,
                     float* __restrict__ out) {
  int i = blockIdx.x * 256 + threadIdx.x;
  if (i >= kB * 2) return;
  int j = i & 1, r = i >> 1;
  float s = b[j];
  for (int c = 0; c < 128; ++c) {
    float v = fmaxf(z[r * 128 + c] * ss[c] + ss[128 + c], 0.0f);
    s += v * W[c * 2 + j];
  }
  out[i] = s;
}

extern "C" void kernel_launch(void* const* d_in, const int* in_sizes, int n_in,
                              void* d_out, int out_size, void* d_ws,
                              size_t ws_size, hipStream_t stream) {
  (void)in_sizes; (void)n_in; (void)out_size; (void)ws_size;
  const float* x = (const float*)d_in[0];
  const int* src = (const int*)d_in[1];  // edge_index row 0 (src); dst == e/16
  const int* batch = (const int*)d_in[3];

  float* ws = (float*)d_ws;
  float* h = ws;                               // N*64
  float* xs = h + (size_t)kN * 64;             // N*192
  float* hagg = xs + (size_t)kN * 192;         // N*64
  float* hjk = hagg;                           // reuse after last conv layer
  float* partE = hagg + (size_t)kN * 64;       // GRID*128
  float* partN = partE + (size_t)GRID_CONV * 128;
  float* ssE = partN + (size_t)GRID_CONV * 128;  // 128
  float* ssN = ssE + 128;                        // 128
  float* ssC = ssN + 128;                        // 256
  float* zbuf = ssC + 256;                       // kB*128
  float* gbuf = zbuf + (size_t)kB * 128;         // kB*64

  for (int L = 0; L < 3; ++L) {
    const float* feat = (L == 0) ? x : h;
    const float* w1 = (const float*)d_in[4 + 6 * L];
    const float* b1 = (const float*)d_in[5 + 6 * L];
    const float* bg = (const float*)d_in[6 + 6 * L];
    const float* bb = (const float*)d_in[7 + 6 * L];
    const float* w2 = (const float*)d_in[8 + 6 * L];
    const float* b2 = (const float*)d_in[9 + 6 * L];
    const float* ng = (const float*)d_in[22 + 2 * L];
    const float* nb = (const float*)d_in[23 + 2 * L];

    if (L == 0)
      conv_passA<7, 16><<<GRID_CONV, 64, 0, stream>>>(feat, src, w1, partE);
    else
      conv_passA<64, 128><<<GRID_CONV, 64, 0, stream>>>(feat, src, w1, partE);
    reduce_stats<<<1, 64, 0, stream>>>(partE, GRID_CONV, 1.0f / kE, b1, bg, bb, ssE);
    if (L == 0)
      conv_passB<7, 16><<<GRID_CONV, 64, 0, stream>>>(feat, src, w1, b1, w2, b2,
                                                      ssE, hagg, partN);
    else
      conv_passB<64, 128><<<GRID_CONV, 64, 0, stream>>>(feat, src, w1, b1, w2,
                                                        b2, ssE, hagg, partN);
    reduce_stats<<<1, 64, 0, stream>>>(partN, GRID_CONV, 1.0f / kN, nullptr, ng,
                                       nb, ssN);
    node_bn_relu<<<(kN * 64) / 256, 256, 0, stream>>>(hagg, ssN, h, xs, 64 * L);
  }

  jk_kernel<<<512, 64, 0, stream>>>(xs, (const float*)d_in[28],
                                    (const float*)d_in[29], hjk);
  pool_kernel<<<kB, 64, 0, stream>>>(hjk, batch, gbuf);
  cls1<<<(kB * 128) / 256, 256, 0, stream>>>(gbuf, (const float*)d_in[30],
                                             (const float*)d_in[31], zbuf);
  cls_stats<<<1, 128, 0, stream>>>(zbuf, (const float*)d_in[32],
                                   (const float*)d_in[33], ssC);
  cls2<<<(kB * 2 + 255) / 256, 256, 0, stream>>>(
      zbuf, ssC, (const float*)d_in[34], (const float*)d_in[35], (float*)d_out);
}